// Normalized_Gradient_Field_weight_46308337385671
// MI455X (gfx1250) — compile-verified
//
#include <hip/hip_runtime.h>
#include <hip/hip_bf16.h>

// ---------------------------------------------------------------------------
// NGF loss, 192^3 fp32 volumes. Memory-bound (85 MB in, 4 B out):
//   - LDS-tiled 18x18x6 halo stencil (edge clamp), async global->LDS gather
//   - per-block 256-way reduction via chained V_WMMA_F32_16X16X4_F32
//   - deterministic two-stage final reduction (no atomics)
// ---------------------------------------------------------------------------

#define AS1 __attribute__((address_space(1)))
#define AS3 __attribute__((address_space(3)))

typedef __attribute__((ext_vector_type(2))) float v2f;
typedef __attribute__((ext_vector_type(8))) float v8f;

#ifndef __has_builtin
#define __has_builtin(x) 0
#endif

#if __has_builtin(__builtin_amdgcn_global_load_async_to_lds_b32)
#define USE_ASYNC_LDS 1
#endif
#if __has_builtin(__builtin_amdgcn_wmma_f32_16x16x4_f32)
#define USE_WMMA_RED 1
#endif

#define NV 192          // volume edge
#define TX 16           // tile x (W)
#define TY 16           // tile y (H)
#define TZ 4            // tile z (D)
#define HX 18           // halo tile extents
#define HY 18
#define HZ 6
#define HTOT (HX*HY*HZ) // 1944 elements per staged array

__device__ __forceinline__ int clampi(int v) {
    return v < 0 ? 0 : (v > NV - 1 ? NV - 1 : v);
}

__global__ __launch_bounds__(256)
void ngf_main(const float* __restrict__ I, const float* __restrict__ J,
              const float* __restrict__ mask, float* __restrict__ ws)
{
    __shared__ float sI[HTOT];
    __shared__ float sJ[HTOT];
    __shared__ float Mred[256];

    const int tid = threadIdx.x;
    const int bx = blockIdx.x % (NV / TX);
    const int by = (blockIdx.x / (NV / TX)) % (NV / TY);
    const int bz = blockIdx.x / ((NV / TX) * (NV / TY));
    const int x0 = bx * TX, y0 = by * TY, z0 = bz * TZ;

    // ---- stage I and J halo tiles into LDS (clamped = edge replication) ----
    for (int i = tid; i < HTOT; i += 256) {
        const int lz  = i / (HX * HY);
        const int rem = i - lz * (HX * HY);
        const int ly  = rem / HX;
        const int lx  = rem - ly * HX;
        const int g = (clampi(z0 - 1 + lz) * NV + clampi(y0 - 1 + ly)) * NV
                      + clampi(x0 - 1 + lx);
#ifdef USE_ASYNC_LDS
        // per-lane gather directly into LDS; tracked by ASYNCcnt
        __builtin_amdgcn_global_load_async_to_lds_b32(
            (AS1 int*)(I + g), (AS3 int*)(sI + i), 0, 0);
        __builtin_amdgcn_global_load_async_to_lds_b32(
            (AS1 int*)(J + g), (AS3 int*)(sJ + i), 0, 0);
#else
        sI[i] = I[g];
        sJ[i] = J[g];
#endif
    }
#ifdef USE_ASYNC_LDS
#if __has_builtin(__builtin_amdgcn_s_wait_asynccnt)
    __builtin_amdgcn_s_wait_asynccnt(0);
#else
    asm volatile("s_wait_asynccnt 0" ::: "memory");
#endif
#endif
    __syncthreads();

    // ---- 18-tap gradients + NGF, 4 voxels per thread (one per z level) ----
    const int tx = tid & 15, ty = tid >> 4;
    float partial = 0.f;
#pragma unroll
    for (int izl = 0; izl < TZ; ++izl) {
        const int cz = izl + 1, cy = ty + 1, cx = tx + 1;
        float Ix = 0.f, Iy = 0.f, Iz = 0.f;
        float Jx = 0.f, Jy = 0.f, Jz = 0.f;
#pragma unroll
        for (int dz = -1; dz <= 1; ++dz) {
#pragma unroll
            for (int dy = -1; dy <= 1; ++dy) {
                const int base = ((cz + dz) * HY + (cy + dy)) * HX + cx;
                const float a  = sI[base - 1], b  = sI[base], c  = sI[base + 1];
                Ix += c - a;
                const float r = a + b + c;
                Iy += (float)dy * r;
                Iz += (float)dz * r;
                const float aj = sJ[base - 1], bj = sJ[base], cj = sJ[base + 1];
                Jx += cj - aj;
                const float rj = aj + bj + cj;
                Jy += (float)dy * rj;
                Jz += (float)dz * rj;
            }
        }
        Ix *= 0.5f; Iy *= 0.5f; Iz *= 0.5f;
        Jx *= 0.5f; Jy *= 0.5f; Jz *= 0.5f;

        const float Imag = Ix * Ix + Iy * Iy + Iz * Iz + 0.01f;  // EPS^2
        const float Jmag = Jx * Jx + Jy * Jy + Jz * Jz + 0.01f;
        const float dt   = Ix * Jx + Iy * Jy + Iz * Jz;
        const float ngf  = (dt * dt) / (Imag * Jmag);

        const int g = ((z0 + izl) * NV + (y0 + ty)) * NV + (x0 + tx);
        partial += (1.0f - ngf) * mask[g];
    }

    // ---- block reduction: 256 partials as a 16x16 matrix, WMMA column-sum ----
    Mred[tid] = partial;
    __syncthreads();

    if (tid < 32) {  // wave 0 only: EXEC all-1s inside this wave (WMMA req.)
        float cs;
#ifdef USE_WMMA_RED
        v2f aA; aA.x = 1.0f; aA.y = 1.0f;       // A = ones(16x4): layout-proof
        v8f d = {0.f, 0.f, 0.f, 0.f, 0.f, 0.f, 0.f, 0.f};
        const int col = tid & 15, hlf = tid >> 4;
#pragma unroll
        for (int c4 = 0; c4 < 4; ++c4) {
            // B chunk = rows 4*c4 .. 4*c4+3 of Mred; with A==ones the column
            // sums are invariant to the exact per-row VGPR placement.
            v2f bB;
            bB.x = Mred[(4 * c4 + hlf) * 16 + col];
            bB.y = Mred[(4 * c4 + 2 + hlf) * 16 + col];
            d = __builtin_amdgcn_wmma_f32_16x16x4_f32(
                false, aA, false, bB, (short)0, d, false, false);
        }
        cs = d[0];  // every row of D holds the 16 column sums
#else
        const int col = tid & 15;
        cs = 0.f;
        if (tid < 16)
            for (int r = 0; r < 16; ++r) cs += Mred[r * 16 + col];
#endif
        // sum the 16 columns (xor stays within each 16-lane half of wave32)
        cs += __shfl_xor(cs, 1);
        cs += __shfl_xor(cs, 2);
        cs += __shfl_xor(cs, 4);
        cs += __shfl_xor(cs, 8);
        if (tid == 0) ws[blockIdx.x] = cs;
    }
}

__global__ __launch_bounds__(256)
void ngf_reduce(const float* __restrict__ ws, float* __restrict__ out, int n)
{
    __shared__ float red[256];
    const int tid = threadIdx.x;
    float s = 0.f;
    for (int i = tid; i < n; i += 256) s += ws[i];   // fixed order per thread
    red[tid] = s;
    __syncthreads();
    for (int ofs = 128; ofs > 0; ofs >>= 1) {        // deterministic tree
        if (tid < ofs) red[tid] += red[tid + ofs];
        __syncthreads();
    }
    if (tid == 0)
        out[0] = red[0] * (1.0f / ((float)NV * (float)NV * (float)NV));
}

extern "C" void kernel_launch(void* const* d_in, const int* in_sizes, int n_in,
                              void* d_out, int out_size, void* d_ws, size_t ws_size,
                              hipStream_t stream)
{
    const float* I    = (const float*)d_in[0];
    const float* J    = (const float*)d_in[1];
    const float* mask = (const float*)d_in[2];
    float* out = (float*)d_out;
    float* ws  = (float*)d_ws;   // needs (192/16)*(192/16)*(192/4)=6912 floats

    const int nblocks = (NV / TX) * (NV / TY) * (NV / TZ);  // 6912
    ngf_main<<<dim3(nblocks), dim3(256), 0, stream>>>(I, J, mask, ws);
    ngf_reduce<<<dim3(1), dim3(256), 0, stream>>>(ws, out, nblocks);

    (void)in_sizes; (void)n_in; (void)out_size; (void)ws_size;
}